// SpectrumEncoding_19937238188590
// MI455X (gfx1250) — compile-verified
//
#include <hip/hip_runtime.h>

typedef __attribute__((ext_vector_type(2))) float v2f;
typedef __attribute__((ext_vector_type(8))) float v8f;

#define D_MODEL 512
#define N_PEAKS 500
#define RESO    10.0f

// One block handles (batch = blockIdx.x>>2, column-quarter = blockIdx.x&3).
// 8 waves per block; each wave owns a 16-column tile and accumulates all 500
// peaks with V_WMMA_F32_16X16X4_F32 (4 peaks per WMMA, K dimension = peaks).
__global__ __launch_bounds__(256)
void spectrum_encoding_wmma(const float* __restrict__ loc_f,
                            const float* __restrict__ inten,
                            const float* __restrict__ pe,
                            float* __restrict__ out)
{
    __shared__ int   s_off[N_PEAKS];   // precomputed row*D_MODEL offsets
    __shared__ float s_w[N_PEAKS];     // peak intensities

    const int b       = blockIdx.x >> 2;
    const int quarter = blockIdx.x & 3;
    const int tid     = threadIdx.x;
    const int wave    = tid >> 5;
    const int lane    = tid & 31;
    const int n       = lane & 15;           // N index within the 16-col tile
    const int kbase   = (lane >> 4) << 1;    // ISA interleave: VGPRi holds rows i, i+2
    const int colbase = quarter * 128 + wave * 16;

    // Stage this batch's indices + weights into LDS once (shared by 8 waves).
    for (int i = tid; i < N_PEAKS; i += 256) {
        float l  = loc_f[b * N_PEAKS + i];
        s_off[i] = ((int)ceilf(l * RESO)) * D_MODEL;   // max 50000*512 < 2^31
        s_w[i]   = inten[b * N_PEAKS + i];
    }
    __syncthreads();

    v8f c = {0.f, 0.f, 0.f, 0.f, 0.f, 0.f, 0.f, 0.f};

    #pragma unroll 5
    for (int p = 0; p < N_PEAKS; p += 4) {
        // A-matrix (16x4): weights broadcast over M. Lane layout:
        //   VGPR0 = w[p+kbase], VGPR1 = w[p+kbase+1]
        const float w0 = s_w[p + kbase];
        const float w1 = s_w[p + kbase + 1];

        // B-matrix (4x16): gathered pe row segments, coalesced 64B per row.
        const int o0 = s_off[p + kbase];
        const int o1 = s_off[p + kbase + 1];
        const float b0 = pe[o0 + colbase + n];
        const float b1 = pe[o1 + colbase + n];

        // Prefetch the next k-group's 4 row segments into the WGP cache
        // (locality=3 -> near-scope prefetch; pe is L2-resident: 102MB < 192MB L2).
        const int pn = (p + 4 < N_PEAKS) ? (p + 4) : p;
        __builtin_prefetch(&pe[s_off[pn + kbase]     + colbase + n], 0, 3);
        __builtin_prefetch(&pe[s_off[pn + kbase + 1] + colbase + n], 0, 3);

        v2f a  = {w0, w1};
        v2f bb = {b0, b1};
        // D = A x B + C ; exact fp32 accumulate on the matrix pipe.
        c = __builtin_amdgcn_wmma_f32_16x16x4_f32(
                /*neg_a=*/false, a, /*neg_b=*/false, bb,
                /*c_mod=*/(short)0, c, /*reuse_a=*/false, /*reuse_b=*/false);
    }

    // Every M row of D holds the same 16 output columns; take M=0 (VGPR0, lanes 0-15).
    if (lane < 16)
        out[b * D_MODEL + colbase + lane] = c[0];
}

extern "C" void kernel_launch(void* const* d_in, const int* in_sizes, int n_in,
                              void* d_out, int out_size, void* d_ws, size_t ws_size,
                              hipStream_t stream) {
    const float* loc_f = (const float*)d_in[0];   // peaks_location  [128,500] f32
    const float* inten = (const float*)d_in[1];   // peaks_intensity [128,500] f32
    const float* pe    = (const float*)d_in[2];   // pe [50001,512] f32
    float*       out   = (float*)d_out;           // [128,512] f32

    // 128 batches * 4 column-quarters, 256 threads (8 wave32) per block.
    spectrum_encoding_wmma<<<dim3(128 * 4), dim3(256), 0, stream>>>(loc_f, inten, pe, out);
}